// JE_12206297055290
// MI455X (gfx1250) — compile-verified
//
#include <hip/hip_runtime.h>
#include <cstdint>
#include <cstddef>

// ---------------------------------------------------------------------------
// Rodrigues rotation, memory-bound streaming kernel for MI455X (gfx1250).
//   in:  px (N,3) f32, qpos_x (N) f32, qpos_y (N) f32, axis (7) f32
//   out: (N,3) f32
// Strategy: ALL input streams staged per-wave through LDS with the CDNA5
// async data mover (global_load_async_to_lds_b128), one s_wait_asynccnt 0,
// compute entirely from LDS (conflict-free), results written in place, then
// contiguous b128 async stores back to global (global_store_async_from_lds).
// ---------------------------------------------------------------------------

#define THREADS      256
#define PTS          8                          // points per thread
#define WAVE_PTS     (32 * PTS)                 // 256 points per wave
#define BLOCK_PTS    (THREADS * PTS)            // 2048 points per block
#define PX_TILE_F    (WAVE_PTS * 3)             // 768 floats (3 KB)
#define Q_TILE_F     (WAVE_PTS)                 // 256 floats (1 KB)
#define WAVE_TILE_F  (PX_TILE_F + 2 * Q_TILE_F) // 1280 floats (5 KB) per wave
#define PX_XFERS     ((PX_TILE_F * 4) / (16 * 32)) // 6 b128 xfers per lane
#define Q_XFERS      ((Q_TILE_F * 4) / (16 * 32))  // 2 b128 xfers per lane

__device__ __forceinline__ void async_load_b128(uint32_t lds_off, const void* gaddr) {
    // copies 16 bytes from global address gaddr to LDS address lds_off   (tracked by ASYNCcnt)
    asm volatile("global_load_async_to_lds_b128 %0, %1, off"
                 :: "v"(lds_off), "v"(gaddr) : "memory");
}

__device__ __forceinline__ void async_store_b128(void* gaddr, uint32_t lds_off) {
    // copies 16 bytes from LDS address lds_off to global address gaddr   (tracked by ASYNCcnt)
    asm volatile("global_store_async_from_lds_b128 %0, %1, off"
                 :: "v"(gaddr), "v"(lds_off) : "memory");
}

__device__ __forceinline__ void wait_asynccnt0() {
    asm volatile("s_wait_asynccnt 0" ::: "memory");
}

__device__ __forceinline__ void wait_dscnt0() {
    asm volatile("s_wait_dscnt 0" ::: "memory");
}

__device__ __forceinline__ void rodrigues_point(
    float x, float y, float z, float theta,
    float dx, float dy, float dz,
    float ox, float oy, float oz,
    float& rx, float& ry, float& rz) {
    float s, c;
    __sincosf(theta, &s, &c);
    // p = px - origin
    float px_ = x - ox, py_ = y - oy, pz_ = z - oz;
    // d . p
    float dp = fmaf(dx, px_, fmaf(dy, py_, dz * pz_));
    // d x p
    float cx = fmaf(dy, pz_, -dz * py_);
    float cy = fmaf(dz, px_, -dx * pz_);
    float cz = fmaf(dx, py_, -dy * px_);
    float k = (1.0f - c) * dp;
    rx = fmaf(c, px_, fmaf(s, cx, fmaf(k, dx, ox)));
    ry = fmaf(c, py_, fmaf(s, cy, fmaf(k, dy, oy)));
    rz = fmaf(c, pz_, fmaf(s, cz, fmaf(k, dz, oz)));
}

__global__ __launch_bounds__(THREADS)
void JE_rodrigues_kernel(const float* __restrict__ px,
                         const float* __restrict__ qx,
                         const float* __restrict__ qy,
                         const float* __restrict__ axis,
                         float* __restrict__ out,
                         int n) {
    __shared__ __align__(16) float tile[(THREADS / 32) * WAVE_TILE_F]; // 40 KB

    const int lane = threadIdx.x & 31;
    const int wave = threadIdx.x >> 5;

    // Uniform parameters (compiler scalarizes -> SMEM + SALU float ops)
    const float ax = axis[0], ay = axis[1], az = axis[2];
    const float ox = axis[3], oy = axis[4], oz = axis[5];
    const float scale = axis[6];
    const float rn = rsqrtf(fmaf(ax, ax, fmaf(ay, ay, az * az)));
    const float dx = ax * rn, dy = ay * rn, dz = az * rn;

    const int wave_base = blockIdx.x * BLOCK_PTS + wave * WAVE_PTS;

    float* wtile  = &tile[wave * WAVE_TILE_F];      // px region (768 f)
    float* qxtile = wtile + PX_TILE_F;              // qx region (256 f)
    float* qytile = qxtile + Q_TILE_F;              // qy region (256 f)
    const uint32_t lds_px = (uint32_t)(uintptr_t)(void*)wtile;
    const uint32_t lds_qx = (uint32_t)(uintptr_t)(void*)qxtile;
    const uint32_t lds_qy = (uint32_t)(uintptr_t)(void*)qytile;

    if (wave_base + WAVE_PTS <= n) {
        // ---------- fast path: whole wave tile in range ----------
        const char* gpx = (const char*)(px + (size_t)wave_base * 3);
        const char* gqx = (const char*)(qx + (size_t)wave_base);
        const char* gqy = (const char*)(qy + (size_t)wave_base);
        char*       gdst = (char*)(out + (size_t)wave_base * 3);

        // Burst-issue ALL async fills (10 b128 per lane), then one wait.
        #pragma unroll
        for (int j = 0; j < PX_XFERS; ++j) {
            const int t = j * 32 + lane;
            async_load_b128(lds_px + (uint32_t)t * 16u, gpx + (size_t)t * 16u);
        }
        #pragma unroll
        for (int j = 0; j < Q_XFERS; ++j) {
            const int t = j * 32 + lane;
            async_load_b128(lds_qx + (uint32_t)t * 16u, gqx + (size_t)t * 16u);
            async_load_b128(lds_qy + (uint32_t)t * 16u, gqy + (size_t)t * 16u);
        }
        wait_asynccnt0();

        // Compute from LDS. px lane-stride 3 dwords, qpos lane-stride 1 dword:
        // both conflict-free across 64 banks. Results overwrite px region.
        #pragma unroll
        for (int j = 0; j < PTS; ++j) {
            const int l = lane + 32 * j;            // local point index
            const float x = wtile[l * 3 + 0];
            const float y = wtile[l * 3 + 1];
            const float z = wtile[l * 3 + 2];
            const float theta = (qytile[l] - qxtile[l]) * scale;
            float rx, ry, rz;
            rodrigues_point(x, y, z, theta, dx, dy, dz, ox, oy, oz, rx, ry, rz);
            wtile[l * 3 + 0] = rx;
            wtile[l * 3 + 1] = ry;
            wtile[l * 3 + 2] = rz;
        }
        wait_dscnt0(); // all this wave's ds_stores visible in LDS

        // Drain: contiguous b128 async stores LDS -> global.
        // S_ENDPGM performs an implicit wait-idle, covering these.
        #pragma unroll
        for (int j = 0; j < PX_XFERS; ++j) {
            const int t = j * 32 + lane;
            async_store_b128(gdst + (size_t)t * 16u, lds_px + (uint32_t)t * 16u);
        }
    } else {
        // ---------- tail fallback: per-point direct global access ----------
        #pragma unroll
        for (int j = 0; j < PTS; ++j) {
            const int p = wave_base + lane + 32 * j;
            if (p < n) {
                const float x = px[(size_t)p * 3 + 0];
                const float y = px[(size_t)p * 3 + 1];
                const float z = px[(size_t)p * 3 + 2];
                const float theta = (qy[p] - qx[p]) * scale;
                float rx, ry, rz;
                rodrigues_point(x, y, z, theta, dx, dy, dz, ox, oy, oz, rx, ry, rz);
                out[(size_t)p * 3 + 0] = rx;
                out[(size_t)p * 3 + 1] = ry;
                out[(size_t)p * 3 + 2] = rz;
            }
        }
    }
}

extern "C" void kernel_launch(void* const* d_in, const int* in_sizes, int n_in,
                              void* d_out, int out_size, void* d_ws, size_t ws_size,
                              hipStream_t stream) {
    (void)n_in; (void)out_size; (void)d_ws; (void)ws_size;
    const float* px   = (const float*)d_in[0]; // (N,3)
    const float* qx   = (const float*)d_in[1]; // (N,)
    const float* qy   = (const float*)d_in[2]; // (N,)
    const float* axis = (const float*)d_in[3]; // (7,)
    float* out = (float*)d_out;

    const int n = in_sizes[1];                 // N from qpos_x
    const int blocks = (n + BLOCK_PTS - 1) / BLOCK_PTS;
    JE_rodrigues_kernel<<<blocks, THREADS, 0, stream>>>(px, qx, qy, axis, out, n);
}